// DisplacementTensors_16003048145210
// MI455X (gfx1250) — compile-verified
//
#include <hip/hip_runtime.h>

typedef float v2f __attribute__((ext_vector_type(2)));
typedef float v8f __attribute__((ext_vector_type(8)));

#define NODES_TOT 20000

// ---------------------------------------------------------------------------
// fp32 WMMA 16x16x4 wrapper (CDNA5: D = A(16x4) * B(4x16) + C, wave32)
// ---------------------------------------------------------------------------
__device__ __forceinline__ v8f wmma_f32(v2f a, v2f b, v8f c) {
  return __builtin_amdgcn_wmma_f32_16x16x4_f32(
      /*neg_a=*/false, a, /*neg_b=*/false, b,
      /*c_mod=*/(short)0, c, /*reuse_a=*/false, /*reuse_b=*/false);
}

// One MLP layer for a 16-row tile, entirely wave-cooperative.
// aBuf: LDS [16][K] row-major activations; w: LDS [K][N]; bias: [N] or null.
// outBuf: LDS [16][N]. col = lane%16, hi = lane/16 (ISA fragment layout).
template <int K, int N, bool RELU>
__device__ __forceinline__ void gemm_stage(const float* aBuf, const float* w,
                                           const float* bias, float* outBuf,
                                           int col, int hi) {
#pragma unroll
  for (int nt = 0; nt < N / 16; ++nt) {
    float bv = bias ? bias[nt * 16 + col] : 0.0f;
    v8f c;
#pragma unroll
    for (int r = 0; r < 8; ++r) c[r] = bv;
#pragma unroll
    for (int k0 = 0; k0 < K; k0 += 4) {
      int k = k0 + 2 * hi;
      v2f a;
      a.x = aBuf[col * K + k];
      a.y = aBuf[col * K + k + 1];
      v2f b;
      b.x = w[k * N + nt * 16 + col];
      b.y = w[(k + 1) * N + nt * 16 + col];
      c = wmma_f32(a, b, c);
    }
#pragma unroll
    for (int r = 0; r < 8; ++r) {
      float v = c[r];
      if (RELU) v = fmaxf(v, 0.1f * v);  // leaky-relu(0.1)
      outBuf[(r + 8 * hi) * N + nt * 16 + col] = v;
    }
  }
}

// ---------------------------------------------------------------------------
// Zero scratch / A_a accumulator (graph-safe, on-stream)
// ---------------------------------------------------------------------------
__global__ void zero_kernel(float* __restrict__ p, long n) {
  long i = (long)blockIdx.x * blockDim.x + threadIdx.x;
  long stride = (long)gridDim.x * blockDim.x;
  for (; i < n; i += stride) p[i] = 0.0f;
}

// ---------------------------------------------------------------------------
// Edge kernel: RBF -> WMMA MLP chain -> atomic scatter (A_a, A_v, A_d sym6)
// 2 waves/block; weights staged to LDS once per block, then a grid-stride
// loop over 16-edge tiles amortizes the staging (~39KB) across many tiles.
// Barrier uniformity: every block runs the same iteration count; per-tile
// validity only gates global loads / atomics, never the barriers or WMMAs.
// ---------------------------------------------------------------------------
#define WAVES_E 2
__global__ __launch_bounds__(64) void edge_kernel(
    const float* __restrict__ r_ij, const float* __restrict__ w_rad,
    const float* __restrict__ b_rad, const float* __restrict__ w_direct,
    const float* __restrict__ w1, const float* __restrict__ b1,
    const float* __restrict__ w2, const float* __restrict__ b2,
    const float* __restrict__ w3, const float* __restrict__ b3,
    const int* __restrict__ edges_src, float* __restrict__ A_a,
    float* __restrict__ Av, float* __restrict__ Ad6, int n_edges, int iters) {
  __shared__ float sWrad[8 * 32];
  __shared__ float sBrad[32];
  __shared__ float sWdir[32 * 32];
  __shared__ float sW1[32 * 64];
  __shared__ float sB1[64];
  __shared__ float sW2[64 * 64];
  __shared__ float sB2[64];
  __shared__ float sW3[64 * 32];
  __shared__ float sB3[32];
  // Per-wave activation staging. sT1's first 128 floats double as the RBF
  // buffer (rbf is dead before t1 is written).
  __shared__ float sHH[WAVES_E][16 * 32];
  __shared__ float sT1[WAVES_E][16 * 64];
  __shared__ float sT2[WAVES_E][16 * 64];
  __shared__ float sRS[WAVES_E][16 * 4];
  __shared__ int sSRC[WAVES_E][16];

  const int tid = threadIdx.x;
  for (int i = tid; i < 8 * 32; i += blockDim.x) sWrad[i] = w_rad[i];
  for (int i = tid; i < 32; i += blockDim.x) sBrad[i] = b_rad[i];
  for (int i = tid; i < 32 * 32; i += blockDim.x) sWdir[i] = w_direct[i];
  for (int i = tid; i < 32 * 64; i += blockDim.x) sW1[i] = w1[i];
  for (int i = tid; i < 64; i += blockDim.x) sB1[i] = b1[i];
  for (int i = tid; i < 64 * 64; i += blockDim.x) sW2[i] = w2[i];
  for (int i = tid; i < 64; i += blockDim.x) sB2[i] = b2[i];
  for (int i = tid; i < 64 * 32; i += blockDim.x) sW3[i] = w3[i];
  for (int i = tid; i < 32; i += blockDim.x) sB3[i] = b3[i];
  __syncthreads();

  const int wave = tid >> 5, lane = tid & 31;
  const int col = lane & 15, hi = lane >> 4;
  const int n_tiles = (n_edges + 15) / 16;
  const int tile_stride = gridDim.x * WAVES_E;

  for (int it = 0; it < iters; ++it) {
    const int tile = (it * gridDim.x + blockIdx.x) * WAVES_E + wave;
    const bool active = tile < n_tiles;
    const int e0 = tile * 16;
    float* rbfBuf = sT1[wave];  // rbf occupies sT1's storage until stage 2

    // Per-edge scalar features (8 Gaussian RBFs + saturated direction rs).
    if (active && lane < 16) {
      int e = e0 + lane;
      // Prefetch next iteration's inputs behind this tile's WMMA chain.
      long enext = (long)(e0 + 16L * tile_stride) + lane;
      if (enext < n_edges) {
        __builtin_prefetch(&r_ij[3 * enext], 0, 1);
        __builtin_prefetch(&edges_src[enext], 0, 1);
      }
      float x = r_ij[3 * e + 0], y = r_ij[3 * e + 1], z = r_ij[3 * e + 2];
      float d = sqrtf(x * x + y * y + z * z);
#pragma unroll
      for (int i = 0; i < 8; ++i) {
        float u = (d - (float)i * (1.0f / 7.0f)) * 8.0f;  // width = R0/8
        rbfBuf[lane * 8 + i] = __expf(-u * u);
      }
      float sx = 7.0f * x, sy = 7.0f * y, sz = 7.0f * z;
      float inv = rsqrtf(1.0f + sx * sx + sy * sy + sz * sz);
      sRS[wave][lane * 4 + 0] = sx * inv;
      sRS[wave][lane * 4 + 1] = sy * inv;
      sRS[wave][lane * 4 + 2] = sz * inv;
      sSRC[wave][lane] = edges_src[e];
    }
    __syncthreads();

    // h = rbf @ w_rad + b_rad                       [16x8]@[8x32]
    gemm_stage<8, 32, false>(rbfBuf, sWrad, sBrad, sHH[wave], col, hi);
    __syncthreads();
    // t1 = lrelu(h @ w1 + b1)                       [16x32]@[32x64]
    gemm_stage<32, 64, true>(sHH[wave], sW1, sB1, sT1[wave], col, hi);
    __syncthreads();
    // t2 = lrelu(t1 @ w2 + b2)                      [16x64]@[64x64]
    gemm_stage<64, 64, true>(sT1[wave], sW2, sB2, sT2[wave], col, hi);
    __syncthreads();

    // rad = h @ w_direct + t2 @ w3 + b3  (fused accumulator), then scatter
    // straight out of the C-fragment layout.
#pragma unroll
    for (int nt = 0; nt < 2; ++nt) {
      float bv = sB3[nt * 16 + col];
      v8f c;
#pragma unroll
      for (int r = 0; r < 8; ++r) c[r] = bv;
#pragma unroll
      for (int k0 = 0; k0 < 32; k0 += 4) {  // + h @ w_direct
        int k = k0 + 2 * hi;
        v2f a;
        a.x = sHH[wave][col * 32 + k];
        a.y = sHH[wave][col * 32 + k + 1];
        v2f b;
        b.x = sWdir[k * 32 + nt * 16 + col];
        b.y = sWdir[(k + 1) * 32 + nt * 16 + col];
        c = wmma_f32(a, b, c);
      }
#pragma unroll
      for (int k0 = 0; k0 < 64; k0 += 4) {  // + t2 @ w3
        int k = k0 + 2 * hi;
        v2f a;
        a.x = sT2[wave][col * 64 + k];
        a.y = sT2[wave][col * 64 + k + 1];
        v2f b;
        b.x = sW3[k * 32 + nt * 16 + col];
        b.y = sW3[(k + 1) * 32 + nt * 16 + col];
        c = wmma_f32(a, b, c);
      }

      if (active) {
        int ach = nt * 16 + col;  // feature channel a
#pragma unroll
        for (int r = 0; r < 8; ++r) {
          int m = r + 8 * hi;  // edge within tile (C layout)
          float val = c[r];
          int src = sSRC[wave][m];
          float rx = sRS[wave][m * 4 + 0];
          float ry = sRS[wave][m * 4 + 1];
          float rz = sRS[wave][m * 4 + 2];
          atomicAdd(&A_a[src * 32 + ach], val);
          float* av = &Av[(src * 32 + ach) * 3];
          atomicAdd(&av[0], val * rx);
          atomicAdd(&av[1], val * ry);
          atomicAdd(&av[2], val * rz);
          float* ad = &Ad6[(src * 32 + ach) * 6];  // symmetric rs⊗rs
          atomicAdd(&ad[0], val * rx * rx);
          atomicAdd(&ad[1], val * rx * ry);
          atomicAdd(&ad[2], val * rx * rz);
          atomicAdd(&ad[3], val * ry * ry);
          atomicAdd(&ad[4], val * ry * rz);
          atomicAdd(&ad[5], val * rz * rz);
        }
      }
    }
    __syncthreads();  // sT1/sT2/sRS reuse safe for next iteration
  }
}

// ---------------------------------------------------------------------------
// Node kernel: out_v = einsum('nac,av'), out_d = einsum('naij,ad') with the
// 6 symmetric components expanded to 9 at store time. WMMA fp32, 8 waves/blk.
// ---------------------------------------------------------------------------
#define WAVES_N 8
__global__ __launch_bounds__(256) void node_kernel(
    const float* __restrict__ Av, const float* __restrict__ Ad6,
    const float* __restrict__ w_v, const float* __restrict__ w_d,
    float* __restrict__ out_v, float* __restrict__ out_d, int n_nodes) {
  __shared__ float sWv[32 * 32];
  __shared__ float sWd[32 * 32];
  for (int i = threadIdx.x; i < 32 * 32; i += blockDim.x) {
    sWv[i] = w_v[i];
    sWd[i] = w_d[i];
  }
  __syncthreads();

  const int wave = threadIdx.x >> 5, lane = threadIdx.x & 31;
  const int col = lane & 15, hi = lane >> 4;
  const int tile = blockIdx.x * WAVES_N + wave;
  if (tile * 16 >= n_nodes) return;
  const int n0 = tile * 16;

  // out_v[n,v,c]: 3 GEMMs [16x32]@[32x32]
#pragma unroll
  for (int cch = 0; cch < 3; ++cch) {
#pragma unroll
    for (int nt = 0; nt < 2; ++nt) {
      v8f acc = {};
#pragma unroll
      for (int k0 = 0; k0 < 32; k0 += 4) {
        int k = k0 + 2 * hi;
        v2f a;
        a.x = Av[((long)(n0 + col) * 32 + k) * 3 + cch];
        a.y = Av[((long)(n0 + col) * 32 + k + 1) * 3 + cch];
        v2f b;
        b.x = sWv[k * 32 + nt * 16 + col];
        b.y = sWv[(k + 1) * 32 + nt * 16 + col];
        acc = wmma_f32(a, b, acc);
      }
#pragma unroll
      for (int r = 0; r < 8; ++r) {
        int node = n0 + r + 8 * hi;
        int v = nt * 16 + col;
        __builtin_nontemporal_store(acc[r],
                                    &out_v[((long)node * 32 + v) * 3 + cch]);
      }
    }
  }

  // out_d[n,d,i,j]: 6 symmetric GEMMs, expand (i,j)/(j,i) at store
  const int si[6] = {0, 0, 0, 1, 1, 2};
  const int sj[6] = {0, 1, 2, 1, 2, 2};
#pragma unroll
  for (int s = 0; s < 6; ++s) {
#pragma unroll
    for (int nt = 0; nt < 2; ++nt) {
      v8f acc = {};
#pragma unroll
      for (int k0 = 0; k0 < 32; k0 += 4) {
        int k = k0 + 2 * hi;
        v2f a;
        a.x = Ad6[((long)(n0 + col) * 32 + k) * 6 + s];
        a.y = Ad6[((long)(n0 + col) * 32 + k + 1) * 6 + s];
        v2f b;
        b.x = sWd[k * 32 + nt * 16 + col];
        b.y = sWd[(k + 1) * 32 + nt * 16 + col];
        acc = wmma_f32(a, b, acc);
      }
#pragma unroll
      for (int r = 0; r < 8; ++r) {
        int node = n0 + r + 8 * hi;
        int dch = nt * 16 + col;
        float val = acc[r];
        float* base = &out_d[((long)node * 32 + dch) * 9];
        __builtin_nontemporal_store(val, &base[si[s] * 3 + sj[s]]);
        if (si[s] != sj[s])
          __builtin_nontemporal_store(val, &base[sj[s] * 3 + si[s]]);
      }
    }
  }
}

// ---------------------------------------------------------------------------
extern "C" void kernel_launch(void* const* d_in, const int* in_sizes, int n_in,
                              void* d_out, int out_size, void* d_ws,
                              size_t ws_size, hipStream_t stream) {
  const float* r_ij = (const float*)d_in[0];
  const float* w_rad = (const float*)d_in[1];
  const float* b_rad = (const float*)d_in[2];
  const float* w_direct = (const float*)d_in[3];
  const float* w1 = (const float*)d_in[4];
  const float* b1 = (const float*)d_in[5];
  const float* w2 = (const float*)d_in[6];
  const float* b2 = (const float*)d_in[7];
  const float* w3 = (const float*)d_in[8];
  const float* b3 = (const float*)d_in[9];
  const float* w_v = (const float*)d_in[10];
  const float* w_d = (const float*)d_in[11];
  const int* edges_src = (const int*)d_in[12];

  const int n_edges = in_sizes[0] / 3;
  const int n_nodes = NODES_TOT;

  float* A_a = (float*)d_out;                     // [N,32]
  float* out_v = A_a + (long)n_nodes * 32;        // [N,32,3]
  float* out_d = out_v + (long)n_nodes * 32 * 3;  // [N,32,3,3]
  float* Av = (float*)d_ws;                       // [N,32,3]
  float* Ad6 = Av + (long)n_nodes * 32 * 3;       // [N,32,6]

  // zero accumulators (A_a + scratch); out_v/out_d fully overwritten
  zero_kernel<<<2048, 256, 0, stream>>>((float*)d_ws, (long)n_nodes * 32 * 9);
  zero_kernel<<<512, 256, 0, stream>>>(A_a, (long)n_nodes * 32);

  // ~16 tiles per wave: weights staged once per block instead of per tile.
  const int n_tiles = (n_edges + 15) / 16;               // 40000
  const int e_blocks = 1250;                             // 2500 waves
  const int tile_stride = e_blocks * WAVES_E;
  const int iters = (n_tiles + tile_stride - 1) / tile_stride;
  edge_kernel<<<e_blocks, 64, 0, stream>>>(r_ij, w_rad, b_rad, w_direct, w1, b1,
                                           w2, b2, w3, b3, edges_src, A_a, Av,
                                           Ad6, n_edges, iters);

  const int nd_tiles = (n_nodes + 15) / 16;
  const int n_blocks = (nd_tiles + WAVES_N - 1) / WAVES_N;
  node_kernel<<<n_blocks, 256, 0, stream>>>(Av, Ad6, w_v, w_d, out_v, out_d,
                                            n_nodes);
}